// ReconStep_63952063037983
// MI455X (gfx1250) — compile-verified
//
#include <hip/hip_runtime.h>

#define NVOX   (128 * 128 * 128)
#define NSTEP  256
#define EPSF   1e-8f

// ---------------------------------------------------------------------------
// Native CDNA5 float atomic: global_atomic_add_f32 (no-return form, executes
// at L2, tracked by STOREcnt). Guaranteed native instruction via inline asm —
// avoids any chance of a CAS-loop expansion from the header helpers.
// ---------------------------------------------------------------------------
__device__ __forceinline__ void atomic_add_f32(float* p, float v) {
#if defined(__gfx1250__)
    asm volatile("global_atomic_add_f32 %0, %1, off"
                 :
                 : "v"(p), "v"(v)
                 : "memory");
#else
    unsafeAtomicAdd(p, v);
#endif
}

// ---------------------------------------------------------------------------
// Zero the accumulation image (d_out is poisoned by the harness).
// ---------------------------------------------------------------------------
__global__ void zero_f4(float4* __restrict__ p, int n4) {
    int i = blockIdx.x * blockDim.x + threadIdx.x;
    if (i < n4) p[i] = make_float4(0.f, 0.f, 0.f, 0.f);
}

// ---------------------------------------------------------------------------
// Compute flat voxel index for sample k along the LOR. Returns in-bounds flag.
// Coordinates precomputed as: c(k) = base + (k+0.5)*step  (units of voxels).
// ---------------------------------------------------------------------------
__device__ __forceinline__ bool sample_vox(float t,
                                           float bx, float by, float bz,
                                           float sx, float sy, float sz,
                                           int& flat) {
    int ix = (int)floorf(fmaf(t, sx, bx));
    int iy = (int)floorf(fmaf(t, sy, by));
    int iz = (int)floorf(fmaf(t, sz, bz));
    // all three in [0,128) iff the OR of their unsigned values is < 128
    unsigned m = (unsigned)ix | (unsigned)iy | (unsigned)iz;
    flat = (ix << 14) | (iy << 7) | iz;
    return m < 128u;
}

// ---------------------------------------------------------------------------
// One thread per LOR. Pass 1: forward project (gather, L2-resident image).
// Pass 2: backproject with run-length-merged float atomics.
// blockIdx.y selects which of the three LOR sets this block processes.
// ---------------------------------------------------------------------------
__global__ __launch_bounds__(256) void trace_mlem(
    const float* __restrict__ img,
    const float* __restrict__ xl,
    const float* __restrict__ yl,
    const float* __restrict__ zl,
    float* __restrict__ back,
    int nlors)
{
    const float* lors = (blockIdx.y == 0) ? xl : (blockIdx.y == 1 ? yl : zl);
    int i = blockIdx.x * blockDim.x + threadIdx.x;
    if (i >= nlors) return;

    const float* L = lors + (size_t)i * 7;
    // Streaming data: non-temporal loads so LOR traffic doesn't evict the
    // L2-resident image (8 MB << 192 MB L2).
    float p1x = __builtin_nontemporal_load(L + 0);
    float p1y = __builtin_nontemporal_load(L + 1);
    float p1z = __builtin_nontemporal_load(L + 2);
    float dx  = __builtin_nontemporal_load(L + 3) - p1x;
    float dy  = __builtin_nontemporal_load(L + 4) - p1y;
    float dz  = __builtin_nontemporal_load(L + 5) - p1z;
    float val = __builtin_nontemporal_load(L + 6);

    const float invvox = 128.0f / 300.0f;   // 1 / (SIZE/GRID)
    const float low    = -150.0f;           // CENTER - SIZE/2
    const float rstep  = 1.0f / (float)NSTEP;

    float bx = (p1x - low) * invvox;
    float by = (p1y - low) * invvox;
    float bz = (p1z - low) * invvox;
    float sx = dx * invvox * rstep;
    float sy = dy * invvox * rstep;
    float sz = dz * invvox * rstep;

    float seg = sqrtf(fmaf(dx, dx, fmaf(dy, dy, dz * dz))) * rstep;

    // ---- Pass 1: forward projection --------------------------------------
    float proj = 0.f;
    int   prev = -1;
    float pval = 0.f;
#pragma unroll 4
    for (int k = 0; k < NSTEP; ++k) {
        int flat;
        if (sample_vox((float)k + 0.5f, bx, by, bz, sx, sy, sz, flat)) {
            if (flat != prev) {            // reuse when consecutive samples
                pval = img[flat];          // hit the same voxel (~2x fewer loads)
                prev = flat;
            }
            proj += pval;
        }
    }
    proj *= seg;
    float w = val / (proj + EPSF) * seg;   // per-sample backprojection weight

    // ---- Pass 2: backprojection with run-length merged atomics -----------
    prev = -1;
    float acc = 0.f;
#pragma unroll 4
    for (int k = 0; k < NSTEP; ++k) {
        int flat;
        if (sample_vox((float)k + 0.5f, bx, by, bz, sx, sy, sz, flat)) {
            if (flat == prev) {
                acc += w;                  // merge consecutive hits: one atomic
            } else {
                if (prev >= 0) atomic_add_f32(back + prev, acc);
                prev = flat;
                acc  = w;
            }
        }
    }
    if (prev >= 0) atomic_add_f32(back + prev, acc);
}

// ---------------------------------------------------------------------------
// Final in-place scale: out = image / (eff + eps) * back   (back lives in out)
// ---------------------------------------------------------------------------
__global__ void combine_f4(const float4* __restrict__ img,
                           const float4* __restrict__ eff,
                           float4* __restrict__ out, int n4) {
    int i = blockIdx.x * blockDim.x + threadIdx.x;
    if (i >= n4) return;
    float4 a = img[i], e = eff[i], b = out[i];
    float4 r;
    r.x = a.x / (e.x + EPSF) * b.x;
    r.y = a.y / (e.y + EPSF) * b.y;
    r.z = a.z / (e.z + EPSF) * b.z;
    r.w = a.w / (e.w + EPSF) * b.w;
    out[i] = r;
}

extern "C" void kernel_launch(void* const* d_in, const int* in_sizes, int n_in,
                              void* d_out, int out_size, void* d_ws, size_t ws_size,
                              hipStream_t stream) {
    const float* img = (const float*)d_in[0];
    const float* eff = (const float*)d_in[1];
    const float* xl  = (const float*)d_in[2];
    const float* yl  = (const float*)d_in[3];
    const float* zl  = (const float*)d_in[4];
    float* out = (float*)d_out;

    const int n4 = NVOX / 4;
    zero_f4<<<(n4 + 255) / 256, 256, 0, stream>>>((float4*)out, n4);

    int nlors = in_sizes[2] / 7;   // 65536 per set
    dim3 grid((nlors + 255) / 256, 3);
    trace_mlem<<<grid, 256, 0, stream>>>(img, xl, yl, zl, out, nlors);

    combine_f4<<<(n4 + 255) / 256, 256, 0, stream>>>(
        (const float4*)img, (const float4*)eff, (float4*)out, n4);
}